// GlobalInitLayer_54305566490875
// MI455X (gfx1250) — compile-verified
//
#include <hip/hip_runtime.h>

#define NUM_GRAPHS 256
#define EDGE_DIM   128
#define GLOBAL_DIM 256
#define LN_EPS     1e-5f
#define ROWS_PER_WAVE 128

typedef __attribute__((ext_vector_type(2))) float v2f;
typedef __attribute__((ext_vector_type(8))) float v8f;

// ---------------------------------------------------------------- zero scratch
__global__ void zero_ws_kernel(float* __restrict__ p, int n) {
    int i = blockIdx.x * blockDim.x + threadIdx.x;
    if (i < n) p[i] = 0.0f;
}

// ---------------------------------------------------------------- segment sum
// Sorted batch ids: accumulate runs in registers, flush atomics on id change.
// One wave handles ROWS_PER_WAVE contiguous rows; 32 lanes x float4 = 128 feats.
__global__ void seg_sum_kernel(const float* __restrict__ edge,
                               const long long* __restrict__ batch,
                               float* __restrict__ acc,
                               float* __restrict__ cnt,
                               int E) {
    const int lane = threadIdx.x & 31;
    const int wid  = (blockIdx.x * blockDim.x + threadIdx.x) >> 5;  // global wave id

    long long r0 = (long long)wid * ROWS_PER_WAVE;
    if (r0 >= E) return;                      // wave-uniform exit
    long long r1 = r0 + ROWS_PER_WAVE;
    if (r1 > E) r1 = E;

    const int f = lane * 4;                   // this lane's 4 features
    float ax = 0.f, ay = 0.f, az = 0.f, aw = 0.f;
    float run = 0.f;
    int   cur = -1;

    for (long long base = r0; base < r1; base += 32) {
        int nb = (int)((r1 - base) < 32 ? (r1 - base) : 32);
        int myid = (lane < nb) ? (int)batch[base + lane] : 0;   // int64 ids fit in int
        for (int j = 0; j < nb; ++j) {
            int id = __shfl(myid, j, 32);
            if (id != cur) {
                if (cur >= 0) {
                    atomicAdd(&acc[cur * EDGE_DIM + f + 0], ax);
                    atomicAdd(&acc[cur * EDGE_DIM + f + 1], ay);
                    atomicAdd(&acc[cur * EDGE_DIM + f + 2], az);
                    atomicAdd(&acc[cur * EDGE_DIM + f + 3], aw);
                    if (lane == 0) atomicAdd(&cnt[cur], run);
                }
                cur = id; ax = ay = az = aw = 0.f; run = 0.f;
            }
            const float4 v = *(const float4*)(edge + (base + j) * (long long)EDGE_DIM + f);
            ax += v.x; ay += v.y; az += v.z; aw += v.w;
            run += 1.f;
        }
    }
    if (cur >= 0) {
        atomicAdd(&acc[cur * EDGE_DIM + f + 0], ax);
        atomicAdd(&acc[cur * EDGE_DIM + f + 1], ay);
        atomicAdd(&acc[cur * EDGE_DIM + f + 2], az);
        atomicAdd(&acc[cur * EDGE_DIM + f + 3], aw);
        if (lane == 0) atomicAdd(&cnt[cur], run);
    }
}

// ---------------------------------------------------------------- mean + GEMM (WMMA f32)
// out tile (16x16) per wave; K loop of 32 x V_WMMA_F32_16X16X4_F32.
// Mean (acc/count) fused into A-fragment load. Bias fused into store.
__global__ void mean_gemm_kernel(const float* __restrict__ acc,
                                 const float* __restrict__ cnt,
                                 const float* __restrict__ W,
                                 const float* __restrict__ bias,
                                 float* __restrict__ ulin) {
    const int lane = threadIdx.x & 31;
    const int wid  = (blockIdx.x * blockDim.x + threadIdx.x) >> 5;
    const int mt   = wid >> 4;      // 0..15 tile row
    const int nt   = wid & 15;      // 0..15 tile col
    const int half = lane >> 4;     // 0: K pair {0,1}, 1: K pair {2,3}
    const int l15  = lane & 15;

    const int   arow = mt * 16 + l15;                       // A-matrix M per lane
    const float inv  = 1.0f / fmaxf(cnt[arow], 1.0f);       // mean scale
    const int   bcol = nt * 16 + l15;                       // B-matrix N per lane

    v8f c = {};
    #pragma unroll 4
    for (int k0 = 0; k0 < EDGE_DIM; k0 += 4) {
        const int ka = k0 + half * 2;
        v2f a, bb;
        a.x  = acc[arow * EDGE_DIM + ka + 0] * inv;
        a.y  = acc[arow * EDGE_DIM + ka + 1] * inv;
        bb.x = W[(ka + 0) * GLOBAL_DIM + bcol];
        bb.y = W[(ka + 1) * GLOBAL_DIM + bcol];
        // 8 args: (neg_a, A, neg_b, B, c_mod, C, reuse_a, reuse_b)
        c = __builtin_amdgcn_wmma_f32_16x16x4_f32(false, a, false, bb,
                                                  (short)0, c, false, false);
    }

    // C/D layout: lanes 0-15 -> M = r, lanes 16-31 -> M = 8 + r; N = lane&15
    const int   n  = nt * 16 + l15;
    const float bn = bias[n];
    #pragma unroll
    for (int r = 0; r < 8; ++r) {
        const int m = mt * 16 + half * 8 + r;
        ulin[m * GLOBAL_DIM + n] = c[r] + bn;
    }
}

// ---------------------------------------------------------------- LayerNorm per row
__global__ void layernorm_kernel(const float* __restrict__ ulin,
                                 const float* __restrict__ lnw,
                                 const float* __restrict__ lnb,
                                 float* __restrict__ out) {
    __shared__ float red[GLOBAL_DIM];
    const int g = blockIdx.x;
    const int t = threadIdx.x;

    const float x = ulin[g * GLOBAL_DIM + t];
    red[t] = x;
    __syncthreads();
    for (int s = GLOBAL_DIM / 2; s > 0; s >>= 1) {
        if (t < s) red[t] += red[t + s];
        __syncthreads();
    }
    const float mu = red[0] * (1.0f / GLOBAL_DIM);
    __syncthreads();

    const float d = x - mu;
    red[t] = d * d;
    __syncthreads();
    for (int s = GLOBAL_DIM / 2; s > 0; s >>= 1) {
        if (t < s) red[t] += red[t + s];
        __syncthreads();
    }
    const float var = red[0] * (1.0f / GLOBAL_DIM);

    out[g * GLOBAL_DIM + t] = d * rsqrtf(var + LN_EPS) * lnw[t] + lnb[t];
}

// ---------------------------------------------------------------- launch
extern "C" void kernel_launch(void* const* d_in, const int* in_sizes, int n_in,
                              void* d_out, int out_size, void* d_ws, size_t ws_size,
                              hipStream_t stream) {
    const float*     edge  = (const float*)d_in[0];      // [E, 128] f32
    const long long* batch = (const long long*)d_in[1];  // [E] int64 (sorted)
    const float*     W     = (const float*)d_in[2];      // [128, 256] f32
    const float*     bias  = (const float*)d_in[3];      // [256] f32
    const float*     lnw   = (const float*)d_in[4];      // [256] f32
    const float*     lnb   = (const float*)d_in[5];      // [256] f32
    float*           out   = (float*)d_out;              // [256, 256] f32

    const int E = in_sizes[1];                            // number of edges

    // workspace layout (floats): acc[256*128] | cnt[256] | ulin[256*256]
    float* acc  = (float*)d_ws;
    float* cnt  = acc + NUM_GRAPHS * EDGE_DIM;
    float* ulin = cnt + NUM_GRAPHS;

    // 1) zero accumulators (ws is poisoned; must re-zero every call)
    const int nz = NUM_GRAPHS * EDGE_DIM + NUM_GRAPHS;
    zero_ws_kernel<<<(nz + 255) / 256, 256, 0, stream>>>(acc, nz);

    // 2) streaming segment-sum (bandwidth-bound stage)
    const int waves  = (E + ROWS_PER_WAVE - 1) / ROWS_PER_WAVE;
    const int blocks = (waves + 7) / 8;                   // 8 waves / 256-thread block
    seg_sum_kernel<<<blocks, 256, 0, stream>>>(edge, batch, acc, cnt, E);

    // 3) mean + 256x128x256 GEMM via v_wmma_f32_16x16x4_f32: 256 tiles = 256 waves
    mean_gemm_kernel<<<32, 256, 0, stream>>>(acc, cnt, W, bias, ulin);

    // 4) per-row LayerNorm
    layernorm_kernel<<<NUM_GRAPHS, GLOBAL_DIM, 0, stream>>>(ulin, lnw, lnb, out);
}